// MPN_53386443489657
// MI455X (gfx1250) — compile-verified
//
#include <hip/hip_runtime.h>
#include <hip/hip_bf16.h>

typedef __attribute__((ext_vector_type(2))) float v2f;
typedef __attribute__((ext_vector_type(8))) float v8f;

#define HDIM 128
#define LDS_PITCH 130   // 128 + 2 pad -> conflict-free ds_load_b64 across 64 banks

// ---------------------------------------------------------------------------
// Scatter-add aggregation: agg[dst] += x[src] for each edge.
// One wave32 per edge; lane handles 4 consecutive floats (float4 gather,
// 4x global_atomic_add_f32 scatter). agg buffer is L2-resident (25.6 MB).
// ---------------------------------------------------------------------------
__global__ void scatter_add_kernel(const float* __restrict__ x,
                                   const int* __restrict__ edge_index,
                                   float* __restrict__ agg,
                                   int E)
{
    const int lane  = threadIdx.x & 31;
    const int wave  = (blockIdx.x * blockDim.x + threadIdx.x) >> 5;
    const int nwave = (gridDim.x * blockDim.x) >> 5;

    for (int e = wave; e < E; e += nwave) {
        const int s = edge_index[e];        // row 0 of edge_index: src
        const int d = edge_index[E + e];    // row 1 of edge_index: dst

        const float4 v = ((const float4*)(x + (size_t)s * HDIM))[lane];
        float* a = agg + (size_t)d * HDIM + lane * 4;
        atomicAdd(a + 0, v.x);
        atomicAdd(a + 1, v.y);
        atomicAdd(a + 2, v.z);
        atomicAdd(a + 3, v.w);
    }
}

// ---------------------------------------------------------------------------
// Fused GraphConv GEMM:  out = [relu]( agg @ W_rel^T + b + xroot @ W_root^T )
// Block: 256 threads = 8 wave32s; block tile = 128 rows x 128 cols.
// Each wave computes a 16x128 strip with 8 x V_WMMA_F32_16X16X4_F32 chains
// over K = 2 x 128.  Both weight matrices staged in dynamic LDS (padded pitch).
// Inner loop is software-pipelined: fragments for k+4 are fetched while the
// 8 WMMAs for k execute, so ds/global latency is hidden behind matrix issue.
//
// Fragment layouts (CDNA5 ISA 7.12.2, 32-bit, wave32):
//   A 16x4 : lane l -> row l%16, VGPR j -> K = 2*(l/16) + j
//   B 4x16 : lane l -> col l%16, VGPR j -> K = 2*(l/16) + j
//   C 16x16: lane l -> col l%16, VGPR v -> row 8*(l/16) + v
// ---------------------------------------------------------------------------
__global__ __launch_bounds__(256) void graphconv_gemm_kernel(
    const float* __restrict__ agg,      // [N,128] aggregated neighbor features
    const float* __restrict__ xroot,    // [N,128] root features
    const float* __restrict__ Wrel,     // [128,128] row-major (torch [out,in])
    const float* __restrict__ Wroot,    // [128,128]
    const float* __restrict__ bias,     // [128]
    float* __restrict__ out,            // [N,128]
    int N, int do_relu)
{
    extern __shared__ float lds[];      // 2 * 128 * LDS_PITCH floats

    const int tid = threadIdx.x;

    // Cooperative weight stage: W[row][k] -> lds[m*128*PITCH + row*PITCH + k]
    for (int i = tid; i < HDIM * (HDIM / 4); i += 256) {
        const int r  = i >> 5;          // weight row (output channel)
        const int c4 = i & 31;          // float4 index within row
        const float4 wr = ((const float4*)(Wrel  + r * HDIM))[c4];
        const float4 wo = ((const float4*)(Wroot + r * HDIM))[c4];
        float* p0 = lds + r * LDS_PITCH + c4 * 4;
        float* p1 = lds + HDIM * LDS_PITCH + r * LDS_PITCH + c4 * 4;
        p0[0] = wr.x; p0[1] = wr.y; p0[2] = wr.z; p0[3] = wr.w;
        p1[0] = wo.x; p1[1] = wo.y; p1[2] = wo.z; p1[3] = wo.w;
    }
    __syncthreads();

    const int waveId = tid >> 5;
    const int lane   = tid & 31;
    const int half   = lane >> 4;       // 0: K pair {0,1}, 1: K pair {2,3}
    const int l16    = lane & 15;

    const int rowBase = blockIdx.x * 128 + waveId * 16;
    const int arow    = min(rowBase + l16, N - 1);   // clamped A-row (EXEC stays full)

    v8f acc[8] = {};

    #pragma unroll
    for (int t = 0; t < 2; ++t) {
        // Per-lane base pointers; fragment (k, j) lives at:
        //   A: Ap[k]                     (2 consecutive floats)
        //   B: Bp[j*16*LDS_PITCH + k]    (2 consecutive floats)
        const float* Ap = (t == 0 ? agg : xroot) + (size_t)arow * HDIM + 2 * half;
        const float* Bp = lds + t * HDIM * LDS_PITCH + l16 * LDS_PITCH + 2 * half;

        v2f a, b[8];

        // Pipeline prologue: fragments for k = 0
        a[0] = Ap[0];
        a[1] = Ap[1];
        #pragma unroll
        for (int j = 0; j < 8; ++j) {
            const float* bp = Bp + j * 16 * LDS_PITCH;
            b[j][0] = bp[0];
            b[j][1] = bp[1];
        }

        // Steady state: fetch k+4 while issuing WMMAs for k
        #pragma unroll 2
        for (int k = 0; k < HDIM - 4; k += 4) {
            v2f an, bn[8];
            const int kn = k + 4;
            an[0] = Ap[kn];
            an[1] = Ap[kn + 1];
            #pragma unroll
            for (int j = 0; j < 8; ++j) {
                const float* bp = Bp + j * 16 * LDS_PITCH + kn;
                bn[j][0] = bp[0];
                bn[j][1] = bp[1];
            }
            #pragma unroll
            for (int j = 0; j < 8; ++j) {
                acc[j] = __builtin_amdgcn_wmma_f32_16x16x4_f32(
                    false, a, false, b[j], (short)0, acc[j], false, false);
            }
            a = an;
            #pragma unroll
            for (int j = 0; j < 8; ++j) b[j] = bn[j];
        }

        // Pipeline epilogue: last k-step
        #pragma unroll
        for (int j = 0; j < 8; ++j) {
            acc[j] = __builtin_amdgcn_wmma_f32_16x16x4_f32(
                false, a, false, b[j], (short)0, acc[j], false, false);
        }
    }

    // Epilogue: bias + optional relu + store. C VGPR v -> row 8*half + v.
    #pragma unroll
    for (int j = 0; j < 8; ++j) {
        const int col = j * 16 + l16;
        const float bc = bias[col];
        #pragma unroll
        for (int v = 0; v < 8; ++v) {
            const int r = rowBase + half * 8 + v;
            if (r < N) {
                float val = acc[j][v] + bc;
                if (do_relu) val = fmaxf(val, 0.0f);
                out[(size_t)r * HDIM + col] = val;
            }
        }
    }
}

// ---------------------------------------------------------------------------
// Host-side launch: 3 layers, each = memset(agg) -> scatter -> fused GEMM.
// ws layout: [ agg : N*128 f32 | h : N*128 f32 ]  (51.2 MB)
// Layer 2 GEMM runs in-place on h (root reads are row-local per wave).
// ---------------------------------------------------------------------------
extern "C" void kernel_launch(void* const* d_in, const int* in_sizes, int n_in,
                              void* d_out, int out_size, void* d_ws, size_t ws_size,
                              hipStream_t stream)
{
    const float* x    = (const float*)d_in[0];
    const int*   ei   = (const int*)d_in[1];
    const float* W1r  = (const float*)d_in[2];
    const float* b1   = (const float*)d_in[3];
    const float* W1o  = (const float*)d_in[4];
    const float* W2r  = (const float*)d_in[5];
    const float* b2   = (const float*)d_in[6];
    const float* W2o  = (const float*)d_in[7];
    const float* W3r  = (const float*)d_in[8];
    const float* b3   = (const float*)d_in[9];
    const float* W3o  = (const float*)d_in[10];

    const int N = in_sizes[0] / HDIM;   // 50000
    const int E = in_sizes[1] / 2;      // 800000

    float* agg = (float*)d_ws;
    float* h   = agg + (size_t)N * HDIM;
    float* y   = (float*)d_out;

    const size_t aggBytes = (size_t)N * HDIM * sizeof(float);
    const size_t ldsBytes = 2u * HDIM * LDS_PITCH * sizeof(float);  // 133120 B (<320KB WGP LDS)

    const dim3 gemmGrid((N + 127) / 128);
    const dim3 scatGrid(8192);
    const dim3 blk(256);

    // ---- layer 1: x -> h (relu) ----
    hipMemsetAsync(agg, 0, aggBytes, stream);
    scatter_add_kernel<<<scatGrid, blk, 0, stream>>>(x, ei, agg, E);
    graphconv_gemm_kernel<<<gemmGrid, blk, ldsBytes, stream>>>(
        agg, x, W1r, W1o, b1, h, N, 1);

    // ---- layer 2: h -> h (relu, in-place) ----
    hipMemsetAsync(agg, 0, aggBytes, stream);
    scatter_add_kernel<<<scatGrid, blk, 0, stream>>>(h, ei, agg, E);
    graphconv_gemm_kernel<<<gemmGrid, blk, ldsBytes, stream>>>(
        agg, h, W2r, W2o, b2, h, N, 1);

    // ---- layer 3: h -> out (no relu) ----
    hipMemsetAsync(agg, 0, aggBytes, stream);
    scatter_add_kernel<<<scatGrid, blk, 0, stream>>>(h, ei, agg, E);
    graphconv_gemm_kernel<<<gemmGrid, blk, ldsBytes, stream>>>(
        agg, h, W3r, W3o, b3, y, N, 0);
}